// TreeSSMReadout_63178968924660
// MI455X (gfx1250) — compile-verified
//
#include <hip/hip_runtime.h>
#include <math.h>
#include <stdint.h>

// Problem constants (match reference)
#define NNODES   16383
#define NPAD     16384
#define D_NODE   512
#define D_SSM    512
#define D_STATE  16
#define ZCOLS    544      // 512 (delta pre-act) + 16 (B) + 16 (C)
#define LN_EPS   1e-5f

typedef __attribute__((ext_vector_type(2))) float v2f;
typedef __attribute__((ext_vector_type(8))) float v8f;

// Async global -> LDS copy (CDNA5 ASYNCcnt path). The low 32 bits of a flat
// pointer into __shared__ are the LDS byte offset.
__device__ __forceinline__ void async_g2l_b128(void* lds, const void* gptr) {
  uint32_t l = (uint32_t)(uintptr_t)lds;
  uint64_t g = (uint64_t)(uintptr_t)gptr;
  asm volatile("global_load_async_to_lds_b128 %0, %1, off" :: "v"(l), "v"(g) : "memory");
}
__device__ __forceinline__ void async_g2l_b32(void* lds, const void* gptr) {
  uint32_t l = (uint32_t)(uintptr_t)lds;
  uint64_t g = (uint64_t)(uintptr_t)gptr;
  asm volatile("global_load_async_to_lds_b32 %0, %1, off" :: "v"(l), "v"(g) : "memory");
}
__device__ __forceinline__ void async_wait0() {
  asm volatile("s_wait_asynccnt 0x0" ::: "memory");
}

// ---------------------------------------------------------------------------
// Prep: logw[n] = log(w[n]+1e-6); Aneg[d,s] = -exp(A_log[d,s])
// ---------------------------------------------------------------------------
__global__ __launch_bounds__(256)
void prep_kernel(const float* __restrict__ w, const float* __restrict__ A_log,
                 float* __restrict__ logw, float* __restrict__ Aneg) {
  int i = blockIdx.x * 256 + threadIdx.x;
  if (i < NNODES) logw[i] = logf(w[i] + 1e-6f);
  if (i < D_SSM * D_STATE) Aneg[i] = -__expf(A_log[i]);
}

// ---------------------------------------------------------------------------
// GEMM1: X[n,j] = sum_k s[n,k]*W_in[j,k]  (K=512)  + logw[n]*W_in[j,512] + b_in[j]
// fp32 WMMA 16x16x4. Block tile 64(M)x32(N), K-step 64, 8 waves = 4x2 subtiles.
// Tiles staged with global_load_async_to_lds_*; fragments prefetched to regs.
// ---------------------------------------------------------------------------
#define G_TM 64
#define G_TN 32
#define G_TK 64

__global__ __launch_bounds__(256)
void gemm1_kernel(const float* __restrict__ S, const float* __restrict__ Win,
                  const float* __restrict__ logw, const float* __restrict__ b_in,
                  float* __restrict__ X) {
  __shared__ float As[G_TM][G_TK + 1];   // 16.6 KB
  __shared__ float Bs[G_TK][G_TN + 1];   //  8.4 KB
  const int tid  = threadIdx.x;
  const int wave = tid >> 5, lane = tid & 31;
  const int wm = wave & 3, wn = wave >> 2;       // 4x2 wave grid
  const int lo = lane & 15, hi = lane >> 4;
  const int blockM = blockIdx.y * G_TM;
  const int blockN = blockIdx.x * G_TN;
  const int fm = wm * 16 + lo;                   // fragment row (A)
  const int fj = wn * 16 + lo;                   // fragment col (B)

  v8f acc = {};
  for (int kb = 0; kb < D_NODE; kb += G_TK) {
    // stage A tile (s rows, stride 512, 16B-aligned) via async B128
#pragma unroll
    for (int i = 0; i < 4; ++i) {
      int q = tid + i * 256;                 // float4 slot in 64x16 grid
      int m = q >> 4, kq = (q & 15) << 2;
      int gm = blockM + m; if (gm > NNODES - 1) gm = NNODES - 1;
      async_g2l_b128(&As[m][kq], S + (size_t)gm * D_NODE + kb + kq);
      if (kb + G_TK < D_NODE)                // exercise global_prefetch_b8
        __builtin_prefetch(S + (size_t)gm * D_NODE + kb + G_TK + kq, 0, 1);
    }
    // stage B tile transposed: Bs[k][j] = Win[blockN+j][kb+k]; row stride 513
    // (not 16B aligned) -> async B32 per element
#pragma unroll
    for (int i = 0; i < 8; ++i) {
      int q = tid + i * 256;
      int j = q >> 6, k = q & 63;
      async_g2l_b32(&Bs[k][j], Win + (size_t)(blockN + j) * (D_NODE + 1) + kb + k);
    }
    async_wait0();
    __syncthreads();
    // prefetch all fragments for this K-step into registers, then WMMA chain
    v2f afr[16], bfr[16];
#pragma unroll
    for (int i = 0; i < 16; ++i) {
      const int k0 = 4 * i + 2 * hi;
      afr[i].x = As[fm][k0];       afr[i].y = As[fm][k0 + 1];
      bfr[i].x = Bs[k0][fj];       bfr[i].y = Bs[k0 + 1][fj];
    }
#pragma unroll
    for (int i = 0; i < 16; ++i)
      acc = __builtin_amdgcn_wmma_f32_16x16x4_f32(false, afr[i], false, bfr[i],
                                                  (short)0, acc, false, false);
    __syncthreads();
  }
  // epilogue: rank-1 logw * W_in[:,512] + bias
  const int gj = blockN + fj;
  const float wlast = Win[(size_t)gj * (D_NODE + 1) + D_NODE];
  const float bj = b_in[gj];
#pragma unroll
  for (int r = 0; r < 8; ++r) {
    int gm = blockM + wm * 16 + hi * 8 + r;
    if (gm < NNODES) X[(size_t)gm * D_SSM + gj] = acc[r] + logw[gm] * wlast + bj;
  }
}

// ---------------------------------------------------------------------------
// GEMM2: Z[n,j] = sum_k X[n,k] * Wcat[j,k], j<512 -> W_delta, <528 -> W_B, else W_C
// (biases / activations applied later in the scan kernel)
// ---------------------------------------------------------------------------
__global__ __launch_bounds__(256)
void gemm2_kernel(const float* __restrict__ X, const float* __restrict__ Wd,
                  const float* __restrict__ Wb, const float* __restrict__ Wc,
                  float* __restrict__ Z) {
  __shared__ float As[G_TM][G_TK + 1];
  __shared__ float Bs[G_TK][G_TN + 1];
  const int tid  = threadIdx.x;
  const int wave = tid >> 5, lane = tid & 31;
  const int wm = wave & 3, wn = wave >> 2;
  const int lo = lane & 15, hi = lane >> 4;
  const int blockM = blockIdx.y * G_TM;
  const int blockN = blockIdx.x * G_TN;
  const int fm = wm * 16 + lo;
  const int fj = wn * 16 + lo;

  v8f acc = {};
  for (int kb = 0; kb < D_SSM; kb += G_TK) {
#pragma unroll
    for (int i = 0; i < 4; ++i) {
      int q = tid + i * 256;
      int m = q >> 4, kq = (q & 15) << 2;
      int gm = blockM + m; if (gm > NNODES - 1) gm = NNODES - 1;
      async_g2l_b128(&As[m][kq], X + (size_t)gm * D_SSM + kb + kq);
    }
    // B tile: concat weight rows, stride 512 (16B aligned) -> 4 async B32 per float4; LDS
    // side is transposed scatter (per-lane LDS offsets are allowed)
#pragma unroll
    for (int i = 0; i < 2; ++i) {
      int q = tid + i * 256;                 // 512 float4 slots = 32 rows x 16
      int j = q >> 4, kq = (q & 15) << 2;
      int gj = blockN + j;
      const float* row = (gj < 512) ? (Wd + (size_t)gj * D_SSM)
                       : (gj < 528) ? (Wb + (size_t)(gj - 512) * D_SSM)
                                    : (Wc + (size_t)(gj - 528) * D_SSM);
      // transposed store needs element granularity -> 4 async B32
#pragma unroll
      for (int e = 0; e < 4; ++e)
        async_g2l_b32(&Bs[kq + e][j], row + kb + kq + e);
    }
    async_wait0();
    __syncthreads();
    v2f afr[16], bfr[16];
#pragma unroll
    for (int i = 0; i < 16; ++i) {
      const int k0 = 4 * i + 2 * hi;
      afr[i].x = As[fm][k0];       afr[i].y = As[fm][k0 + 1];
      bfr[i].x = Bs[k0][fj];       bfr[i].y = Bs[k0 + 1][fj];
    }
#pragma unroll
    for (int i = 0; i < 16; ++i)
      acc = __builtin_amdgcn_wmma_f32_16x16x4_f32(false, afr[i], false, bfr[i],
                                                  (short)0, acc, false, false);
    __syncthreads();
  }
  const int gj = blockN + fj;
#pragma unroll
  for (int r = 0; r < 8; ++r) {
    int gm = blockM + wm * 16 + hi * 8 + r;
    if (gm < NNODES) Z[(size_t)gm * ZCOLS + gj] = acc[r];
  }
}

// ---------------------------------------------------------------------------
// Tree scan, one BFS level per launch, one block per node (256 thr, 2 rows/thr).
// H[n,d,s] = exp(delta*A)*H[par,d,s] + delta*x*B[s]; Y, LayerNorm fused.
// Leaf H (level 13) is never stored (halves the H traffic).
// ---------------------------------------------------------------------------
__global__ __launch_bounds__(256)
void scan_level_kernel(const float* __restrict__ Z, const float* __restrict__ X,
                       const float* __restrict__ logw,
                       const float* __restrict__ b_delta, const float* __restrict__ W_w,
                       const float* __restrict__ b_w, const float* __restrict__ b_B,
                       const float* __restrict__ b_C, const float* __restrict__ Aneg,
                       const float* __restrict__ Dv, const float* __restrict__ gamma,
                       const float* __restrict__ beta,
                       float* __restrict__ H, float* __restrict__ out,
                       int levelStart, int hasParent, int writeH) {
  __shared__ float Bsh[D_STATE], Csh[D_STATE];
  __shared__ float red1[256], red2[256];
  const int tid = threadIdx.x;
  const int n   = levelStart + blockIdx.x;
  const size_t zrow = (size_t)n * ZCOLS;
  if (tid < D_STATE) {
    Bsh[tid] = Z[zrow + 512 + tid] + b_B[tid];
    Csh[tid] = Z[zrow + 528 + tid] + b_C[tid];
  }
  __syncthreads();
  const float lw  = logw[n];
  const int   par = (n - 1) >> 1;
  float yv[2];
#pragma unroll
  for (int half = 0; half < 2; ++half) {
    const int d = tid + half * 256;
    const float pre   = Z[zrow + d] + b_delta[d];
    const float sp    = (pre > 20.f) ? pre : log1pf(__expf(pre));
    const float sg    = 1.f / (1.f + __expf(-(lw * W_w[d] + b_w[d])));
    const float delta = sp * sg;
    const float xd    = X[(size_t)n * D_SSM + d];
    const float dx    = delta * xd;
    float hp[D_STATE], ar[D_STATE], hs[D_STATE];
    if (hasParent) {
      const float4* hp4 = reinterpret_cast<const float4*>(H + (size_t)par * (D_SSM * D_STATE) + d * D_STATE);
#pragma unroll
      for (int q = 0; q < 4; ++q) {
        float4 v = hp4[q];
        hp[4 * q + 0] = v.x; hp[4 * q + 1] = v.y; hp[4 * q + 2] = v.z; hp[4 * q + 3] = v.w;
      }
    } else {
#pragma unroll
      for (int s = 0; s < D_STATE; ++s) hp[s] = 0.f;
    }
    const float4* a4 = reinterpret_cast<const float4*>(Aneg + d * D_STATE);
#pragma unroll
    for (int q = 0; q < 4; ++q) {
      float4 v = a4[q];
      ar[4 * q + 0] = v.x; ar[4 * q + 1] = v.y; ar[4 * q + 2] = v.z; ar[4 * q + 3] = v.w;
    }
    float y = 0.f;
#pragma unroll
    for (int s = 0; s < D_STATE; ++s) {
      float h = __expf(delta * ar[s]) * hp[s] + dx * Bsh[s];
      hs[s] = h;
      y += h * Csh[s];
    }
    if (writeH) {
      float4* ho = reinterpret_cast<float4*>(H + (size_t)n * (D_SSM * D_STATE) + d * D_STATE);
#pragma unroll
      for (int q = 0; q < 4; ++q)
        ho[q] = make_float4(hs[4 * q + 0], hs[4 * q + 1], hs[4 * q + 2], hs[4 * q + 3]);
    }
    yv[half] = y + Dv[d] * xd;
  }
  // fused LayerNorm over 512 rows
  red1[tid] = yv[0] + yv[1];
  red2[tid] = yv[0] * yv[0] + yv[1] * yv[1];
  __syncthreads();
  for (int s = 128; s > 0; s >>= 1) {
    if (tid < s) { red1[tid] += red1[tid + s]; red2[tid] += red2[tid + s]; }
    __syncthreads();
  }
  const float mean = red1[0] * (1.f / 512.f);
  const float var  = red2[0] * (1.f / 512.f) - mean * mean;
  const float rstd = rsqrtf(var + LN_EPS);
#pragma unroll
  for (int half = 0; half < 2; ++half) {
    const int d = tid + half * 256;
    out[(size_t)n * D_SSM + d] = (yv[half] - mean) * rstd * gamma[d] + beta[d];
  }
}

// ---------------------------------------------------------------------------
extern "C" void kernel_launch(void* const* d_in, const int* in_sizes, int n_in,
                              void* d_out, int out_size, void* d_ws, size_t ws_size,
                              hipStream_t stream) {
  (void)in_sizes; (void)n_in; (void)out_size; (void)ws_size;
  const float* s       = (const float*)d_in[0];
  const float* w       = (const float*)d_in[1];
  // d_in[2]=parent, d_in[3]=depth: complete binary tree in BFS order -> implicit
  const float* W_in    = (const float*)d_in[4];
  const float* b_in    = (const float*)d_in[5];
  const float* W_delta = (const float*)d_in[6];
  const float* b_delta = (const float*)d_in[7];
  const float* W_w     = (const float*)d_in[8];
  const float* b_w     = (const float*)d_in[9];
  const float* A_log   = (const float*)d_in[10];
  const float* Dv      = (const float*)d_in[11];
  const float* W_B     = (const float*)d_in[12];
  const float* b_B     = (const float*)d_in[13];
  const float* W_C     = (const float*)d_in[14];
  const float* b_C     = (const float*)d_in[15];
  const float* gamma   = (const float*)d_in[16];
  const float* beta    = (const float*)d_in[17];
  float* out = (float*)d_out;

  // workspace layout (~338 MB):
  //   Aneg  32 KB | logw 64 KB | X 32 MB | Z 34 MB | H (internal nodes only) 256 MB
  char*  ws   = (char*)d_ws;
  float* Aneg = (float*)ws;                                   ws += (size_t)D_SSM * D_STATE * 4;
  float* logw = (float*)ws;                                   ws += (size_t)NPAD * 4;
  float* X    = (float*)ws;                                   ws += (size_t)NPAD * D_SSM * 4;
  float* Zb   = (float*)ws;                                   ws += (size_t)NPAD * ZCOLS * 4;
  float* H    = (float*)ws;   // 8191 * 512 * 16 floats

  prep_kernel<<<64, 256, 0, stream>>>(w, A_log, logw, Aneg);

  dim3 g1(D_SSM / G_TN, NPAD / G_TM);     // (16, 256)
  gemm1_kernel<<<g1, 256, 0, stream>>>(s, W_in, logw, b_in, X);

  dim3 g2(ZCOLS / G_TN, NPAD / G_TM);     // (17, 256)
  gemm2_kernel<<<g2, 256, 0, stream>>>(X, W_delta, W_B, W_C, Zb);

  for (int l = 0; l < 14; ++l) {
    scan_level_kernel<<<dim3(1u << l), 256, 0, stream>>>(
        Zb, X, logw, b_delta, W_w, b_w, b_B, b_C, Aneg, Dv, gamma, beta,
        H, out, (1 << l) - 1, (l > 0) ? 1 : 0, (l < 13) ? 1 : 0);
  }
}